// GreedyFeatureInit_75350906241373
// MI455X (gfx1250) — compile-verified
//
#include <hip/hip_runtime.h>
#include <hip/hip_bf16.h>

#define HW   32
#define NPAT 1024        // HW*HW
#define DIM  768
#define NB   32          // batch
#define NSLOT 8

typedef __attribute__((ext_vector_type(2))) float    v2f;
typedef __attribute__((ext_vector_type(8))) float    v8f;
typedef __attribute__((ext_vector_type(4))) unsigned v4u;
typedef __attribute__((ext_vector_type(8))) int      v8i;
typedef __attribute__((ext_vector_type(4))) int      v4i;

// ---------------- K1: inverse row norms ----------------
__global__ void __launch_bounds__(256)
k_norms(const float* __restrict__ f, float* __restrict__ inv_norm) {
    const int row = blockIdx.x;                 // b*NPAT + p
    const float* fr = f + (size_t)row * DIM;
    float ss = 0.f;
    for (int d = threadIdx.x; d < DIM; d += 256) { float v = fr[d]; ss += v * v; }
    __shared__ float red[256];
    red[threadIdx.x] = ss; __syncthreads();
    for (int s = 128; s > 0; s >>= 1) {
        if (threadIdx.x < s) red[threadIdx.x] += red[threadIdx.x + s];
        __syncthreads();
    }
    if (threadIdx.x == 0)
        inv_norm[row] = 1.0f / fmaxf(sqrtf(red[0]), 1e-12f);
}

// ---------------- K2: normalized global mean per batch ----------------
__global__ void __launch_bounds__(256)
k_gmean(const float* __restrict__ f, float* __restrict__ gmean_n) {
    const int b = blockIdx.x;
    const float* fb = f + (size_t)b * NPAT * DIM;
    __shared__ float gm[DIM];
    for (int d = threadIdx.x; d < DIM; d += 256) {
        float s = 0.f;
        for (int p = 0; p < NPAT; ++p) s += fb[(size_t)p * DIM + d];
        gm[d] = s * (1.0f / (float)NPAT);
    }
    __syncthreads();
    float ss = 0.f;
    for (int d = threadIdx.x; d < DIM; d += 256) ss += gm[d] * gm[d];
    __shared__ float red[256];
    __shared__ float s_inv;
    red[threadIdx.x] = ss; __syncthreads();
    for (int s = 128; s > 0; s >>= 1) {
        if (threadIdx.x < s) red[threadIdx.x] += red[threadIdx.x + s];
        __syncthreads();
    }
    if (threadIdx.x == 0) s_inv = 1.0f / fmaxf(sqrtf(red[0]), 1e-12f);
    __syncthreads();
    const float inv = s_inv;
    for (int d = threadIdx.x; d < DIM; d += 256)
        gmean_n[(size_t)b * DIM + d] = gm[d] * inv;
}

// ---------------- K3: saliency = local_sim - global_sim ----------------
__global__ void __launch_bounds__(256)
k_saliency(const float* __restrict__ f, const float* __restrict__ inv_norm,
           const float* __restrict__ gmean_n, float* __restrict__ saliency) {
    const int b = blockIdx.x >> 10;             // /NPAT
    const int p = blockIdx.x & (NPAT - 1);
    const int h = p >> 5, w = p & 31;

    // uniform neighbor list (zero padding => skip OOB)
    int   nrow[9]; float ninv[9]; int cnt = 0;
    for (int dh = -1; dh <= 1; ++dh)
        for (int dw = -1; dw <= 1; ++dw) {
            int hh = h + dh, ww = w + dw;
            if (hh < 0 || hh >= HW || ww < 0 || ww >= HW) continue;
            int np = hh * HW + ww;
            nrow[cnt] = np;
            ninv[cnt] = inv_norm[b * NPAT + np];
            ++cnt;
        }
    const float invc = inv_norm[b * NPAT + p];
    const float* fb = f + (size_t)b * NPAT * DIM;
    const float* fc = fb + (size_t)p * DIM;
    const float* gm = gmean_n + (size_t)b * DIM;

    float s_loc = 0.f, s_glob = 0.f;
    for (int d = threadIdx.x; d < DIM; d += 256) {
        float cn = fc[d] * invc;
        float acc = 0.f;
        for (int i = 0; i < cnt; ++i)
            acc += fb[(size_t)nrow[i] * DIM + d] * ninv[i];
        s_loc  += cn * acc;
        s_glob += cn * gm[d];
    }
    __shared__ float r0[256], r1[256];
    r0[threadIdx.x] = s_loc; r1[threadIdx.x] = s_glob; __syncthreads();
    for (int s = 128; s > 0; s >>= 1) {
        if (threadIdx.x < s) { r0[threadIdx.x] += r0[threadIdx.x + s];
                               r1[threadIdx.x] += r1[threadIdx.x + s]; }
        __syncthreads();
    }
    if (threadIdx.x == 0)
        saliency[blockIdx.x] = r0[0] * (1.0f / 9.0f) - r1[0];
}

// ---------------- K4: greedy selection (one block per batch) ----------------
// Raw-dot formulation: dot(fn[m], fn[idx]) = invn[m]*invn[idx]*dot(f[m], f[idx]),
// so WMMA operates directly on raw features (no per-element scaling in the loop)
// and the two scalar norms are applied to the 16 accumulated dots at the end.
//   V_WMMA_F32_16X16X4_F32 tiling:
//     A = 16 patches x 4 k (lanes 0-15: K=0,1 ; lanes 16-31: K=2,3)
//     B = sel k-slice replicated in all 16 columns -> every column is the dot
//     D:  lane 0 holds M=0..7 (col 0), lane 16 holds M=8..15
// The per-iteration `sel` staging is a verbatim 3KB row copy features[b,idx,:]
// -> LDS, done with the Tensor Data Mover (tensor_load_to_lds + TENSORcnt).
__global__ void __launch_bounds__(256)
k_greedy(const float* __restrict__ f, const float* __restrict__ inv_norm,
         const float* __restrict__ saliency_g, float* __restrict__ out) {
    const int b   = blockIdx.x;
    const int tid = threadIdx.x;
    const int lane = tid & 31, wave = tid >> 5;

    __shared__ float sal[NPAT];
    __shared__ float msim[NPAT];
    __shared__ float invn[NPAT];
    __shared__ float sel[DIM];          // raw selected row, TDM-staged
    __shared__ float redv[256];
    __shared__ int   redi[256];
    __shared__ int   s_idx;

    for (int i = tid; i < NPAT; i += 256) {
        sal[i]  = saliency_g[b * NPAT + i];
        msim[i] = 0.f;
        invn[i] = inv_norm[b * NPAT + i];
    }
    __syncthreads();

    const float* fb = f + (size_t)b * NPAT * DIM;

    for (int s = 0; s < NSLOT; ++s) {
        // ---- argmax over score = sal - msim (first index wins ties) ----
        float bv = -3.0e38f; int bi = 0;
        for (int i = tid; i < NPAT; i += 256) {
            float sc = sal[i] - msim[i];
            if (sc > bv) { bv = sc; bi = i; }
        }
        redv[tid] = bv; redi[tid] = bi; __syncthreads();
        for (int r = 128; r > 0; r >>= 1) {
            if (tid < r) {
                float v2 = redv[tid + r]; int i2 = redi[tid + r];
                if (v2 > redv[tid] || (v2 == redv[tid] && i2 < redi[tid])) {
                    redv[tid] = v2; redi[tid] = i2;
                }
            }
            __syncthreads();
        }
        if (tid == 0) s_idx = redi[0];
        __syncthreads();
        const int idx = s_idx;
        const int h = idx >> 5, w = idx & 31;
        const int h0 = max(h - 1, 0), h1 = min(h + 1, HW - 1);
        const int w0 = max(w - 1, 0), w1 = min(w + 1, HW - 1);
        const float rcount = 1.0f / (float)((h1 - h0 + 1) * (w1 - w0 + 1));
        const float isel = invn[idx];

        // ---- slot = window mean of RAW features ----
        for (int d = tid; d < DIM; d += 256) {
            float sum = 0.f;
            for (int hh = h0; hh <= h1; ++hh)
                for (int ww = w0; ww <= w1; ++ww)
                    sum += fb[(size_t)(hh * HW + ww) * DIM + d];
            out[((size_t)b * NSLOT + s) * DIM + d] = sum * rcount;
        }

        // ---- TDM: copy raw row features[b, idx, :] (3KB) into LDS `sel` ----
        if (wave == 0) {
            unsigned long long ga =
                (unsigned long long)(const void*)(fb + (size_t)idx * DIM);
            unsigned lds_off = (unsigned)(size_t)(void*)sel;   // flat->LDS low 32b
            v4u g0;
            g0.x = 1u;                                          // count=1, user mode
            g0.y = lds_off;                                     // lds_addr [63:32]
            g0.z = (unsigned)(ga & 0xFFFFFFFFu);                // global_addr [95:64]
            g0.w = (unsigned)((ga >> 32) & 0x01FFFFFFu)         // global_addr [120:96]
                 | (2u << 30);                                  // type=2 (image)
            v8i g1;
            g1[0] = 0x20000;                  // data_size=2 -> 4-byte elements
            g1[1] = (int)(DIM << 16);         // tensor_dim0[15:0] @ bits 63:48
            g1[2] = (int)(1u << 16);          // dim0 hi=0 | tensor_dim1=1 @ 95:80
            g1[3] = (int)(DIM << 16);         // dim1 hi=0 | tile_dim0=768 @ 127:112
            g1[4] = 1;                        // tile_dim1=1 (tile_dim2=0)
            g1[5] = DIM;                      // tensor_dim0_stride low 32
            g1[6] = 0;                        // stride hi / dim1_stride
            g1[7] = 0;
            v4i gz4 = {0, 0, 0, 0};           // groups 2/3 unused (<=2D tensor)
            v8i gz8 = {0, 0, 0, 0, 0, 0, 0, 0};
            __builtin_amdgcn_tensor_load_to_lds(g0, g1, gz4, gz4, gz8, 0);
            __builtin_amdgcn_s_wait_tensorcnt(0);
        }
        __syncthreads();

        // ---- WMMA f32 raw dots: 8 waves x 8 tiles x 16 patches = 1024 ----
        const int half = lane >> 4;                  // 0: K=0,1  1: K=2,3
        for (int t = wave * 8; t < wave * 8 + 8; ++t) {
            const int m0 = t * 16;
            const int m  = m0 + (lane & 15);
            const float* arow = fb + (size_t)m * DIM;
            v8f c = {};
            #pragma unroll 2
            for (int k0 = 0; k0 < DIM; k0 += 4) {
                const int ko = k0 + 2 * half;
                float2 av  = *reinterpret_cast<const float2*>(arow + ko);
                float2 bv2 = *reinterpret_cast<const float2*>(&sel[ko]);
                v2f a;  a.x  = av.x;  a.y  = av.y;
                v2f bb; bb.x = bv2.x; bb.y = bv2.y;
                c = __builtin_amdgcn_wmma_f32_16x16x4_f32(
                        false, a, false, bb, (short)0, c, false, false);
            }
            if (lane == 0) {
                #pragma unroll
                for (int j = 0; j < 8; ++j)
                    msim[m0 + j] = fmaxf(msim[m0 + j], c[j] * invn[m0 + j] * isel);
            } else if (lane == 16) {
                #pragma unroll
                for (int j = 0; j < 8; ++j)
                    msim[m0 + 8 + j] =
                        fmaxf(msim[m0 + 8 + j], c[j] * invn[m0 + 8 + j] * isel);
            }
        }
        __syncthreads();
        if (tid == 0) sal[idx] = -1e30f;             // mark selected
        __syncthreads();
    }
}

extern "C" void kernel_launch(void* const* d_in, const int* in_sizes, int n_in,
                              void* d_out, int out_size, void* d_ws, size_t ws_size,
                              hipStream_t stream) {
    const float* f = (const float*)d_in[0];          // [32, 1024, 768] fp32
    float* out = (float*)d_out;                      // [32, 8, 768] fp32
    float* ws = (float*)d_ws;

    float* inv_norm = ws;                            // 32768
    float* gmean_n  = ws + NB * NPAT;                // 24576
    float* saliency = gmean_n + NB * DIM;            // 32768

    k_norms<<<NB * NPAT, 256, 0, stream>>>(f, inv_norm);
    k_gmean<<<NB, 256, 0, stream>>>(f, gmean_n);
    k_saliency<<<NB * NPAT, 256, 0, stream>>>(f, inv_norm, gmean_n, saliency);
    k_greedy<<<NB, 256, 0, stream>>>(f, inv_norm, saliency, out);
}